// TemporalAttention_23871428232071
// MI455X (gfx1250) — compile-verified
//
#include <hip/hip_runtime.h>
#include <hip/hip_bf16.h>
#include <math.h>

// ---------------------------------------------------------------------------
// Problem constants (match reference)
// ---------------------------------------------------------------------------
#define BQ   2
#define SQ   2048
#define DQ   1024
#define HQ   16
#define DKQ  64
#define SCALEQ 0.125f          // 1/sqrt(64)
#define DECAY_INV (1.0f/0.9f)
#define VWEIGHT 0.1f
#define LDSW 2052              // 2048 + 4 pad -> row stride % 64 banks == 4
#define PARTW 17               // padded row stride for phase-3 partial tiles
#define PART_TILE (16 * PARTW) // 272 floats per 16x16 partial tile

typedef float v2f __attribute__((ext_vector_type(2)));
typedef float v8f __attribute__((ext_vector_type(8)));

// Full-precision fp32 matrix core: V_WMMA_F32_16X16X4_F32
// args: (neg_a, A, neg_b, B, c_mod, C, reuse_a, reuse_b)
static __device__ __forceinline__ v8f wmma4(v2f a, v2f b, v8f c) {
    return __builtin_amdgcn_wmma_f32_16x16x4_f32(false, a, false, b, (short)0, c,
                                                 false, false);
}

static __device__ __forceinline__ v2f ld2(const float* p) {
    const float2 t = *(const float2*)p;
    v2f r; r.x = t.x; r.y = t.y;
    return r;
}

// ---------------------------------------------------------------------------
// Kernel 1: fused Q/K/V projections. grid.z in {0,1,2} selects q/k/v.
// Each wave computes a 64x16 tile (4 M sub-tiles sharing every B load):
// per K-step: 4 coalesced A b64 loads + 2 strided B b32 loads + 4 wmma.
// Output scattered into [B,H,S,64] layout for the attention kernel.
// ---------------------------------------------------------------------------
__global__ void qkv_proj_kernel(const float* __restrict__ q,
                                const float* __restrict__ k,
                                const float* __restrict__ v,
                                const float* __restrict__ Wq, const float* __restrict__ bq,
                                const float* __restrict__ Wk, const float* __restrict__ bk,
                                const float* __restrict__ Wv, const float* __restrict__ bv,
                                float* __restrict__ Qb, float* __restrict__ Kb,
                                float* __restrict__ Vb) {
    const int z = blockIdx.z;
    const float* X    = (z == 0) ? q  : (z == 1) ? k  : v;
    const float* W    = (z == 0) ? Wq : (z == 1) ? Wk : Wv;
    const float* bias = (z == 0) ? bq : (z == 1) ? bk : bv;
    float*       Out  = (z == 0) ? Qb : (z == 1) ? Kb : Vb;

    const int wave = threadIdx.x >> 5;
    const int lane = threadIdx.x & 31;
    const int m0 = blockIdx.y * 64;
    const int n0 = (blockIdx.x * 8 + wave) * 16;
    const int mr = lane & 15;
    const int half = lane >> 4;
    const int koff = half * 2;

    const float* a0 = X + (size_t)(m0 + mr) * DQ + koff;
    const float* wp = W + n0 + mr;

    v8f acc[4] = {{}, {}, {}, {}};
    for (int kk0 = 0; kk0 < DQ; kk0 += 16) {
        if (kk0 + 64 < DQ) {
            __builtin_prefetch(wp + (size_t)(kk0 + 64 + koff) * DQ, 0, 3);
            __builtin_prefetch(a0 + kk0 + 64, 0, 3);
        }
#pragma unroll
        for (int kk = kk0; kk < kk0 + 16; kk += 4) {
            v2f b;
            b.x = wp[(size_t)(kk + koff)     * DQ];
            b.y = wp[(size_t)(kk + koff + 1) * DQ];
#pragma unroll
            for (int s = 0; s < 4; ++s)
                acc[s] = wmma4(ld2(a0 + (size_t)s * 16 * DQ + kk), b, acc[s]);
        }
    }

    const int ncol = n0 + mr;
    const int hh = ncol >> 6, dk = ncol & 63;
    const float bval = bias[ncol];
#pragma unroll
    for (int s = 0; s < 4; ++s) {
#pragma unroll
        for (int vv = 0; vv < 8; ++vv) {
            const int row = m0 + s * 16 + vv + 8 * half;   // [0,4096)
            const int bb = row >> 11, ss = row & (SQ - 1);
            Out[((size_t)(bb * HQ + hh) * SQ + ss) * DKQ + dk] = acc[s][vv] + bval;
        }
    }
}

// ---------------------------------------------------------------------------
// Kernel 2: generic GEMM + bias (+optional relu). C[M,N] = A[M,K]@W[K,N]+b.
// Same 64x16-per-wave register blocking as above.
// ---------------------------------------------------------------------------
__global__ void gemm_bias_kernel(const float* __restrict__ A,
                                 const float* __restrict__ W,
                                 const float* __restrict__ bias,
                                 float* __restrict__ Cmat,
                                 int K, int N, int relu) {
    const int wave = threadIdx.x >> 5;
    const int lane = threadIdx.x & 31;
    const int m0 = blockIdx.y * 64;
    const int n0 = (blockIdx.x * 8 + wave) * 16;
    const int mr = lane & 15;
    const int half = lane >> 4;
    const int koff = half * 2;

    const float* a0 = A + (size_t)(m0 + mr) * K + koff;
    const float* wp = W + n0 + mr;

    v8f acc[4] = {{}, {}, {}, {}};
    for (int kk0 = 0; kk0 < K; kk0 += 16) {
        if (kk0 + 64 < K) {
            __builtin_prefetch(wp + (size_t)(kk0 + 64 + koff) * N, 0, 3);
            __builtin_prefetch(a0 + kk0 + 64, 0, 3);
        }
#pragma unroll
        for (int kk = kk0; kk < kk0 + 16; kk += 4) {
            v2f b;
            b.x = wp[(size_t)(kk + koff)     * N];
            b.y = wp[(size_t)(kk + koff + 1) * N];
#pragma unroll
            for (int s = 0; s < 4; ++s)
                acc[s] = wmma4(ld2(a0 + (size_t)s * 16 * K + kk), b, acc[s]);
        }
    }

    const float bval = bias[n0 + mr];
#pragma unroll
    for (int s = 0; s < 4; ++s) {
#pragma unroll
        for (int vv = 0; vv < 8; ++vv) {
            const int row = m0 + s * 16 + vv + 8 * half;
            float val = acc[s][vv] + bval;
            if (relu) val = fmaxf(val, 0.0f);
            Cmat[(size_t)row * N + n0 + mr] = val;
        }
    }
}

// ---------------------------------------------------------------------------
// Kernel 3: conf[r] = sigmoid(dot(H[r,0:512], Wc2) + bc2). One wave per row.
// ---------------------------------------------------------------------------
__global__ void conf_reduce_kernel(const float* __restrict__ Hbuf,
                                   const float* __restrict__ Wc2,
                                   const float* __restrict__ bc2,
                                   float* __restrict__ conf) {
    const int r = blockIdx.x * 8 + (threadIdx.x >> 5);
    const int lane = threadIdx.x & 31;
    const float* hrow = Hbuf + (size_t)r * (DQ / 2);
    float s = 0.0f;
    for (int j = lane; j < DQ / 2; j += 32) s += hrow[j] * Wc2[j];
    for (int off = 16; off > 0; off >>= 1) s += __shfl_xor(s, off, 32);
    if (lane == 0) conf[r] = 1.0f / (1.0f + expf(-(s + bc2[0])));
}

// ---------------------------------------------------------------------------
// Kernel 4: fused attention. One workgroup (8 waves) = one (b,h) x 16 query
// rows. Full 16x2048 score stripe lives in LDS (~128KB of the 320KB WGP
// LDS). Phase1: WMMA QK^T + biases -> LDS. Phase2: row softmax in LDS,
// write attn once. Phase3: P(16x2048) @ V(2048x64) with A from LDS,
// K split across all 8 waves, partials combined via padded LDS tiles.
// ---------------------------------------------------------------------------
__global__ void fused_attn_kernel(const float* __restrict__ Qb,
                                  const float* __restrict__ Kb,
                                  const float* __restrict__ Vb,
                                  const float* __restrict__ td,
                                  const float* __restrict__ conf,
                                  float* __restrict__ attn,
                                  float* __restrict__ ctx) {
    extern __shared__ float sc[];   // [16][LDSW] scores + [4][PART_TILE] partials

    const int blk     = blockIdx.x;
    const int row_blk = blk & (SQ / 16 - 1);     // 0..127
    const int bh      = blk >> 7;                // 0..31
    const int b       = bh >> 4;
    const int h       = bh & (HQ - 1);
    const int row0    = row_blk * 16;

    const int wave = threadIdx.x >> 5;
    const int lane = threadIdx.x & 31;
    const int mr   = lane & 15;
    const int half = lane >> 4;
    const int koff = half * 2;

    const float* Qp  = Qb + ((size_t)bh * SQ + row0) * DKQ;
    const float* Kp  = Kb + (size_t)bh * SQ * DKQ;
    const float* Vp  = Vb + (size_t)bh * SQ * DKQ;
    const float* tdb = td + (size_t)b * SQ;

    // ---- Phase 1: scores = Q K^T * scale + temporal_bias + conf_bias ----
    v2f areg[16];
#pragma unroll
    for (int kk = 0; kk < 16; ++kk)
        areg[kk] = ld2(Qp + (size_t)mr * DKQ + kk * 4 + koff);

    float dtq[8], cb[8];
#pragma unroll
    for (int vv = 0; vv < 8; ++vv) {
        const int row = row0 + vv + 8 * half;
        dtq[vv] = tdb[row];
        cb[vv]  = conf[(size_t)b * SQ + row] * VWEIGHT;
    }

    for (int t8 = wave; t8 < SQ / 16; t8 += 8) {
        const int n0 = t8 * 16;
        if (t8 + 8 < SQ / 16)
            __builtin_prefetch(Kp + (size_t)(n0 + 128 + mr) * DKQ + koff, 0, 3);
        // two interleaved accumulators to break the WMMA RAW chain
        v8f accA = {}, accB = {};
#pragma unroll
        for (int kk = 0; kk < 16; kk += 2) {
            const float* krow0 = Kp + (size_t)(n0 + mr) * DKQ + koff;
            accA = wmma4(areg[kk],     ld2(krow0 + kk * 4),     accA);
            accB = wmma4(areg[kk + 1], ld2(krow0 + kk * 4 + 4), accB);
        }
        const v8f acc = accA + accB;
        const float dtk = tdb[n0 + mr];
#pragma unroll
        for (int vv = 0; vv < 8; ++vv) {
            float sval = acc[vv] * SCALEQ;
            const float d = fabsf(dtq[vv] - dtk);
            sval += logf(expf(-d * DECAY_INV) + 1e-8f) + cb[vv];
            sc[(size_t)(vv + 8 * half) * LDSW + n0 + mr] = sval;
        }
    }
    __syncthreads();

    // ---- Phase 2: row softmax in LDS; write attn to global exactly once ----
    for (int r = wave * 2; r < wave * 2 + 2; ++r) {
        float* row = sc + (size_t)r * LDSW;
        float mx = -3.4e38f;
        for (int j = lane; j < SQ; j += 32) mx = fmaxf(mx, row[j]);
        for (int off = 16; off > 0; off >>= 1) mx = fmaxf(mx, __shfl_xor(mx, off, 32));
        float sum = 0.0f;
        for (int j = lane; j < SQ; j += 32) {
            const float e = expf(row[j] - mx);
            row[j] = e; sum += e;
        }
        for (int off = 16; off > 0; off >>= 1) sum += __shfl_xor(sum, off, 32);
        const float inv = 1.0f / sum;
        float* ag = attn + ((size_t)bh * SQ + row0 + r) * SQ;
        for (int j = lane; j < SQ; j += 32) {
            const float p = row[j] * inv;
            row[j] = p;
            ag[j] = p;
        }
    }
    __syncthreads();

    // ---- Phase 3: ctx = P(16x2048) @ V(2048x64); all 8 waves active:
    //      wave = (tile 0..3) x (K-half 0..1); partials merged via LDS. ----
    {
        const int t  = wave & 3;
        const int kh = wave >> 2;
        const int n0 = t * 16;
        const int kbeg = kh * (SQ / 2), kend = kbeg + SQ / 2;
        v8f accA = {}, accB = {};
        for (int kk0 = kbeg; kk0 < kend; kk0 += 32) {
            if (kk0 + 64 < kend)
                __builtin_prefetch(Vp + (size_t)(kk0 + 64 + koff) * DKQ + n0 + mr, 0, 3);
#pragma unroll
            for (int kk = kk0; kk < kk0 + 32; kk += 8) {
                const int krow = kk + koff;
                v2f b0, b1;
                b0.x = Vp[(size_t)krow       * DKQ + n0 + mr];
                b0.y = Vp[(size_t)(krow + 1) * DKQ + n0 + mr];
                b1.x = Vp[(size_t)(krow + 4) * DKQ + n0 + mr];
                b1.y = Vp[(size_t)(krow + 5) * DKQ + n0 + mr];
                accA = wmma4(ld2(sc + (size_t)mr * LDSW + krow),     b0, accA);
                accB = wmma4(ld2(sc + (size_t)mr * LDSW + krow + 4), b1, accB);
            }
        }
        const v8f acc = accA + accB;
        float* part = sc + 16 * LDSW;
        if (kh == 1) {
#pragma unroll
            for (int vv = 0; vv < 8; ++vv)
                part[t * PART_TILE + (vv + 8 * half) * PARTW + mr] = acc[vv];
        }
        __syncthreads();
        if (kh == 0) {
#pragma unroll
            for (int vv = 0; vv < 8; ++vv) {
                const int row = row0 + vv + 8 * half;
                ctx[((size_t)b * SQ + row) * DQ + h * DKQ + n0 + mr] =
                    acc[vv] + part[t * PART_TILE + (vv + 8 * half) * PARTW + mr];
            }
        }
    }
}

// ---------------------------------------------------------------------------
// Host-side launcher
// ---------------------------------------------------------------------------
extern "C" void kernel_launch(void* const* d_in, const int* in_sizes, int n_in,
                              void* d_out, int out_size, void* d_ws, size_t ws_size,
                              hipStream_t stream) {
    (void)in_sizes; (void)n_in; (void)out_size; (void)ws_size;

    const float* query = (const float*)d_in[0];
    const float* key   = (const float*)d_in[1];
    const float* value = (const float*)d_in[2];
    const float* tdel  = (const float*)d_in[3];
    const float* Wq    = (const float*)d_in[4];
    const float* bq    = (const float*)d_in[5];
    const float* Wk    = (const float*)d_in[6];
    const float* bk    = (const float*)d_in[7];
    const float* Wv    = (const float*)d_in[8];
    const float* bv    = (const float*)d_in[9];
    const float* Wo    = (const float*)d_in[10];
    const float* bo    = (const float*)d_in[11];
    const float* Wc1   = (const float*)d_in[12];
    const float* bc1   = (const float*)d_in[13];
    const float* Wc2   = (const float*)d_in[14];
    const float* bc2   = (const float*)d_in[15];

    // workspace layout (floats)
    const size_t proj_elems = (size_t)BQ * HQ * SQ * DKQ;   // 4,194,304
    float* ws   = (float*)d_ws;
    float* Qb   = ws;
    float* Kb   = Qb + proj_elems;
    float* Vb   = Kb + proj_elems;
    float* Hbuf = Vb + proj_elems;                          // [4096, 512]
    float* conf = Hbuf + (size_t)BQ * SQ * (DQ / 2);        // [4096]
    float* ctx  = conf + (size_t)BQ * SQ;                   // [4096, 1024]

    float* outp = (float*)d_out;                            // [B,S,D]
    float* attn = outp + (size_t)BQ * SQ * DQ;              // [B,H,S,S]

    const int lds_bytes = (16 * LDSW + 4 * PART_TILE) * (int)sizeof(float);
    hipFuncSetAttribute((const void*)fused_attn_kernel,
                        hipFuncAttributeMaxDynamicSharedMemorySize, lds_bytes);

    // 1) Q/K/V projections (64x16 per-wave tiles)
    qkv_proj_kernel<<<dim3(DQ / 128, (BQ * SQ) / 64, 3), 256, 0, stream>>>(
        query, key, value, Wq, bq, Wk, bk, Wv, bv, Qb, Kb, Vb);

    // 2) confidence hidden layer: H = relu(query @ Wc1 + bc1), N = 512
    gemm_bias_kernel<<<dim3((DQ / 2) / 128, (BQ * SQ) / 64), 256, 0, stream>>>(
        query, Wc1, bc1, Hbuf, DQ, DQ / 2, 1);

    // 3) confidence scalar per row
    conf_reduce_kernel<<<dim3((BQ * SQ) / 8), 256, 0, stream>>>(Hbuf, Wc2, bc2, conf);

    // 4) fused scores + bias + softmax + attn-write + P@V
    fused_attn_kernel<<<dim3(BQ * HQ * (SQ / 16)), 256, lds_bytes, stream>>>(
        Qb, Kb, Vb, tdel, conf, attn, ctx);

    // 5) output projection: out = ctx @ Wo + bo
    gemm_bias_kernel<<<dim3(DQ / 128, (BQ * SQ) / 64), 256, 0, stream>>>(
        ctx, Wo, bo, outp, DQ, DQ, 0);
}